// RepeatEncoder_34969623724326
// MI455X (gfx1250) — compile-verified
//
#include <hip/hip_runtime.h>

#ifndef __has_builtin
#define __has_builtin(x) 0
#endif

#define AS1 __attribute__((address_space(1)))
#define AS3 __attribute__((address_space(3)))

typedef int   v4i __attribute__((ext_vector_type(4)));
typedef float v4f __attribute__((ext_vector_type(4)));

// Problem constants (fixed by the reference)
constexpr int B_DIM   = 64;
constexpr int C_DIM   = 64;
constexpr int L_DIM   = 512;
constexpr int T_STEPS = 32;

constexpr int TILE_L    = 32;          // l-slice per block
constexpr int THREADS   = 256;         // 8 wave32
constexpr int ROWSTRIDE = 36;          // floats per LDS row (144 B: 16B-aligned, bank-friendly)

// LIF with decay_input=True, tau=2, v_th=1, hard reset to 0:
//   v = v + (x - v)/2 = 0.5*(v + x);  s = (v >= 1);  v = s ? 0 : v
__global__ __launch_bounds__(THREADS)
void lif_repeat_encoder(const float* __restrict__ in, float* __restrict__ out) {
    __shared__ float tile[C_DIM * ROWSTRIDE + 4];   // [c][l] with padded rows, ~9 KB

    const int tid = threadIdx.x;
    const int b   = blockIdx.x >> 4;                 // L/TILE_L = 16 tiles per batch
    const int l0  = (blockIdx.x & 15) * TILE_L;

    // ---------------- Phase 1: async global -> LDS staging (transpose source) ----------
    // Tile is 64 rows (channels) x 128 B (32 floats of l). 512 x 16B lane-loads total,
    // i.e. 2 async b128 loads per thread. Reads are fully coalesced along l.
    const float* src = in + (size_t)b * (C_DIM * L_DIM) + l0;
#pragma unroll
    for (int i = 0; i < 2; ++i) {
        const int s = tid + i * THREADS;             // 0..511
        const int c = s >> 3;                        // channel row, 0..63
        const int k = (s & 7) << 2;                  // float offset within row: 0,4,...,28
        const float* gp = src + (size_t)c * L_DIM + k;
        float*       lp = &tile[c * ROWSTRIDE + k];
#if __has_builtin(__builtin_amdgcn_global_load_async_to_lds_b128)
        __builtin_amdgcn_global_load_async_to_lds_b128((AS1 v4i*)gp, (AS3 v4i*)lp,
                                                       /*offset=*/0, /*cpol=*/0);
#else
        const unsigned laddr = (unsigned)(size_t)(AS3 float*)lp;  // 32-bit LDS address
        const unsigned long long gaddr = (unsigned long long)gp;
        asm volatile("global_load_async_to_lds_b128 %0, %1, off"
                     :: "v"(laddr), "v"(gaddr) : "memory");
#endif
    }
#if __has_builtin(__builtin_amdgcn_s_wait_asynccnt)
    __builtin_amdgcn_s_wait_asynccnt(0);
#else
    asm volatile("s_wait_asynccnt 0" ::: "memory");
#endif
    __syncthreads();

    // ---------------- Phase 2: 32-step LIF recurrence + streaming NT stores ------------
    // Thread -> 4 consecutive channels at one l; wave writes contiguous 128B+ per step.
    const int c4 = (tid & 15) << 2;                  // 0,4,...,60
    const int lb = tid >> 4;                         // 0..15
    float* outb = out + (size_t)b * (T_STEPS * L_DIM * C_DIM);

#pragma unroll
    for (int pass = 0; pass < 2; ++pass) {
        const int l = lb + pass * 16;                // 0..31 within tile
        const float x0 = tile[(c4 + 0) * ROWSTRIDE + l];
        const float x1 = tile[(c4 + 1) * ROWSTRIDE + l];
        const float x2 = tile[(c4 + 2) * ROWSTRIDE + l];
        const float x3 = tile[(c4 + 3) * ROWSTRIDE + l];

        float v0 = 0.0f, v1 = 0.0f, v2 = 0.0f, v3 = 0.0f;
        // out[b][t][l0+l][c4..c4+3]
        v4f* dst = (v4f*)(outb + (size_t)(l0 + l) * C_DIM + c4);

#pragma unroll
        for (int t = 0; t < T_STEPS; ++t) {
            v0 = 0.5f * (v0 + x0);
            v1 = 0.5f * (v1 + x1);
            v2 = 0.5f * (v2 + x2);
            v3 = 0.5f * (v3 + x3);
            v4f o;
            o.x = (v0 >= 1.0f) ? 1.0f : 0.0f;
            o.y = (v1 >= 1.0f) ? 1.0f : 0.0f;
            o.z = (v2 >= 1.0f) ? 1.0f : 0.0f;
            o.w = (v3 >= 1.0f) ? 1.0f : 0.0f;
            __builtin_nontemporal_store(o, dst);     // output (256MB) > L2 (192MB): stream it
            dst += (L_DIM * C_DIM) / 4;              // advance one timestep (128 KB)
            v0 = (v0 >= 1.0f) ? 0.0f : v0;           // hard reset
            v1 = (v1 >= 1.0f) ? 0.0f : v1;
            v2 = (v2 >= 1.0f) ? 0.0f : v2;
            v3 = (v3 >= 1.0f) ? 0.0f : v3;
        }
    }
}

extern "C" void kernel_launch(void* const* d_in, const int* in_sizes, int n_in,
                              void* d_out, int out_size, void* d_ws, size_t ws_size,
                              hipStream_t stream) {
    (void)in_sizes; (void)n_in; (void)out_size; (void)d_ws; (void)ws_size;
    const float* in = (const float*)d_in[0];
    float*      out = (float*)d_out;
    const int nblocks = B_DIM * (L_DIM / TILE_L);    // 1024
    lif_repeat_encoder<<<nblocks, THREADS, 0, stream>>>(in, out);
}